// PatchAbstractionMSG_15857019257415
// MI455X (gfx1250) — compile-verified
//
#include <hip/hip_runtime.h>

// ---------------------------------------------------------------------------
// PatchAbstractionMSG for MI455X (gfx1250, wave32).
// FPS + KNN are latency-bound scalar kernels; the MLP/residual GEMMs use
// V_WMMA_F32_16X16X4_F32 (full f32 precision; GEMM FLOPs are not the
// bottleneck here, so no precision is sacrificed for rate).
// ---------------------------------------------------------------------------

#define B_    8
#define N_    16384
#define S_    1024
#define KMAX  32
#define OUTC  259
#define BN_EPS 1e-5f

typedef __attribute__((ext_vector_type(2))) float v2f;
typedef __attribute__((ext_vector_type(8))) float v8f;

struct MlpParams {
  const float *w0, *b0, *g0, *be0, *mu0, *va0;
  const float *w1, *b1, *g1, *be1, *mu1, *va1;
  const float *w2, *b2, *g2, *be2, *mu2, *va2;
  const float *rw1, *rb1, *rg1, *rbe1, *rmu1, *rva1;
  const float *rw2, *rb2, *rg2, *rbe2, *rmu2, *rva2;
};

__device__ __forceinline__ void wave_argmax(float& v, int& i) {
#pragma unroll
  for (int off = 16; off > 0; off >>= 1) {
    float ov = __shfl_xor(v, off);
    int   oi = __shfl_xor(i, off);
    if (ov > v || (ov == v && oi < i)) { v = ov; i = oi; }
  }
}

__device__ __forceinline__ void wave_argmin(float& v, int& i) {
#pragma unroll
  for (int off = 16; off > 0; off >>= 1) {
    float ov = __shfl_xor(v, off);
    int   oi = __shfl_xor(i, off);
    if (ov < v || (ov == v && oi < i)) { v = ov; i = oi; }
  }
}

// ---------------------------------------------------------------------------
// 1) Farthest point sampling: one block per batch, 1024 threads, 16 points
//    per thread. dist residues live in VGPRs; x stays hot in L2 (1.5 MB).
// ---------------------------------------------------------------------------
__global__ __launch_bounds__(1024) void fps_kernel(
    const float* __restrict__ x, float* __restrict__ cent,
    float* __restrict__ out)
{
  const int b    = blockIdx.x;
  const int tid  = threadIdx.x;
  const int lane = tid & 31;
  const int wv   = __builtin_amdgcn_readfirstlane(tid >> 5);

  __shared__ int   s_cur;
  __shared__ float rv[32];
  __shared__ int   ri[32];

  float dist[16];
#pragma unroll
  for (int i = 0; i < 16; ++i) dist[i] = 1e10f;

  if (tid == 0) s_cur = 0;
  __syncthreads();

  const float* xb = x + (size_t)b * N_ * 3;

  for (int s = 0; s < S_; ++s) {
    const int cur = s_cur;
    const float cx = xb[cur * 3 + 0];
    const float cy = xb[cur * 3 + 1];
    const float cz = xb[cur * 3 + 2];

    if (tid == 0) {
      const size_t cs = (size_t)b * S_ + s;
      cent[cs * 3 + 0] = cx; cent[cs * 3 + 1] = cy; cent[cs * 3 + 2] = cz;
      out[cs * OUTC + 0] = cx; out[cs * OUTC + 1] = cy; out[cs * OUTC + 2] = cz;
    }

    float bv = -1.0f; int bi = 0;
#pragma unroll
    for (int i = 0; i < 16; ++i) {
      const int p = i * 1024 + tid;
      const float dx = xb[p * 3 + 0] - cx;
      const float dy = xb[p * 3 + 1] - cy;
      const float dz = xb[p * 3 + 2] - cz;
      const float dd = dx * dx + dy * dy + dz * dz;
      const float d  = fminf(dist[i], dd);
      dist[i] = d;
      if (d > bv) { bv = d; bi = p; }   // ascending p => first-max tie-break
    }
    wave_argmax(bv, bi);
    if (lane == 0) { rv[wv] = bv; ri[wv] = bi; }
    __syncthreads();
    if (wv == 0) {
      bv = rv[lane]; bi = ri[lane];
      wave_argmax(bv, bi);
      if (lane == 0) s_cur = bi;
    }
    __syncthreads();
  }
}

// ---------------------------------------------------------------------------
// 2) KNN top-32 (ascending distance): one wave per centroid, 8 waves/block.
//    Per-lane local top-32 lists live in LDS (conflict-free slot striding),
//    merged with 32 rounds of wave-argmin extraction.
// ---------------------------------------------------------------------------
__global__ __launch_bounds__(256) void knn_kernel(
    const float* __restrict__ x, const float* __restrict__ cent,
    int* __restrict__ knn)
{
  __shared__ float sd[KMAX * 256];
  __shared__ int   si[KMAX * 256];

  const int tid  = threadIdx.x;
  const int lane = tid & 31;
  const int wv   = __builtin_amdgcn_readfirstlane(tid >> 5);
  const int cs   = blockIdx.x * 8 + wv;   // flat centroid id, 0..B*S-1
  const int b    = cs >> 10;              // / S_

  const float cx = cent[cs * 3 + 0];
  const float cy = cent[cs * 3 + 1];
  const float cz = cent[cs * 3 + 2];
  const float* xb = x + (size_t)b * N_ * 3;
  const int slot = tid;                   // per-lane list column in LDS

  float worst = -1.0f; int wpos = 0;
#pragma unroll
  for (int e = 0; e < KMAX; ++e) {        // seed with first 32 points per lane
    const int p = e * 32 + lane;
    const float dx = xb[p * 3 + 0] - cx;
    const float dy = xb[p * 3 + 1] - cy;
    const float dz = xb[p * 3 + 2] - cz;
    const float d = dx * dx + dy * dy + dz * dz;
    sd[e * 256 + slot] = d; si[e * 256 + slot] = p;
    if (d > worst) { worst = d; wpos = e; }
  }
  for (int p = KMAX * 32 + lane; p < N_; p += 32) {
    const float dx = xb[p * 3 + 0] - cx;
    const float dy = xb[p * 3 + 1] - cy;
    const float dz = xb[p * 3 + 2] - cz;
    const float d = dx * dx + dy * dy + dz * dz;
    if (d < worst) {
      sd[wpos * 256 + slot] = d; si[wpos * 256 + slot] = p;
      worst = -1.0f;
      for (int e = 0; e < KMAX; ++e) {
        const float de = sd[e * 256 + slot];
        if (de > worst) { worst = de; wpos = e; }
      }
    }
  }

  // Merge: 32 extraction rounds -> ascending distance order (matches top_k)
  for (int r = 0; r < KMAX; ++r) {
    float mv = 1e30f; int mi = 0x7fffffff; int mpos = 0;
    for (int e = 0; e < KMAX; ++e) {
      const float de = sd[e * 256 + slot];
      const int   ie = si[e * 256 + slot];
      if (de < mv || (de == mv && ie < mi)) { mv = de; mi = ie; mpos = e; }
    }
    float bv = mv; int bi = mi;
    wave_argmin(bv, bi);
    if (lane == 0) knn[(size_t)cs * KMAX + r] = bi;
    if (bi == mi) sd[mpos * 256 + slot] = 1e30f;  // owner marks entry consumed
  }
}

// ---------------------------------------------------------------------------
// 3) WMMA tile GEMM + fused bias/BN/ReLU epilogue, fully unrolled.
//    NM x NN 16x16 tiles distributed over 8 waves (NM*NN % 8 == 0), K-loop
//    unrolled in steps of 4. A tile 16xK from LDS (row-major, padded
//    stride); B tile Kx16 from the (out,in)-major weight matrix
//    (B[k][n] = W[n*ldw + k]). Fragment layouts per CDNA5 ISA Sec 7.12.2:
//      A: lane%16 = M, element j -> K = j + 2*(lane/16)
//      B: lane%16 = N, element j -> K = j + 2*(lane/16)
//      C/D: lane%16 = N, element i -> M = i + 8*(lane/16)
// ---------------------------------------------------------------------------
template <int NM, int NN, int KPAD>
__device__ __forceinline__ void gemm_bn_relu(
    const float* __restrict__ As, int lda,
    const float* __restrict__ W, int ldw,
    const float* __restrict__ bias,
    const float* __restrict__ gm, const float* __restrict__ bt,
    const float* __restrict__ mu, const float* __restrict__ va,
    float* __restrict__ Ds, int ldd,
    int wv, int lane)
{
  const int rowSel = lane & 15;
  const int kSel   = (lane >> 4) << 1;
  constexpr int NIT = (NM * NN) / 8;     // tiles per wave (compile-time)
#pragma unroll
  for (int it = 0; it < NIT; ++it) {
    const int t  = wv + it * 8;          // wv is wave-uniform (readfirstlane)
    const int mt = t / NN;               // NN is a power of two -> shifts
    const int nt = t % NN;
    v8f acc = {};
    const float* ap = As + (mt * 16 + rowSel) * lda + kSel;
    const float* bp = W + (nt * 16 + rowSel) * ldw + kSel;
#pragma unroll
    for (int k0 = 0; k0 < KPAD; k0 += 4) {
      v2f a, bm;
      a[0]  = ap[k0 + 0];
      a[1]  = ap[k0 + 1];
      bm[0] = bp[k0 + 0];
      bm[1] = bp[k0 + 1];
      acc = __builtin_amdgcn_wmma_f32_16x16x4_f32(
          false, a, false, bm, (short)0, acc, false, false);
    }
    const int n = nt * 16 + rowSel;      // N is per-lane constant in C/D layout
    const float sc = gm[n] * rsqrtf(va[n] + BN_EPS);
    const float sh = bt[n] - mu[n] * sc;
    const float bb = bias[n];
    const int mr = mt * 16 + ((lane >> 4) << 3);
#pragma unroll
    for (int i = 0; i < 8; ++i) {
      const float v = (acc[i] + bb) * sc + sh;
      Ds[(mr + i) * ldd + n] = fmaxf(v, 0.0f);
    }
  }
}

// ---------------------------------------------------------------------------
// 4) Per-scale MLP + max-pool + residual. Block = 64 points (NC centroids),
//    256 threads = 8 waves. LDS strides padded (68/132) => conflict-free
//    column loads for A fragments (stride % 64 banks == 4).
// ---------------------------------------------------------------------------
#define LD1 68    // padded stride for 64-wide activations
#define LD2 132   // padded stride for 128-wide activations

template <int KP>
__global__ __launch_bounds__(256) void mlp_kernel(
    const float* __restrict__ x, const float* __restrict__ cent,
    const int* __restrict__ knn, MlpParams P,
    float* __restrict__ out, int cbase)
{
  constexpr int NC = 64 / KP;           // centroids per block
  __shared__ float s_a[64 * LD1];       // 17 KB  (gather / act2 / h+r1)
  __shared__ float s_b[64 * LD2];       // 33 KB  (act1 / act3)
  __shared__ float s_w0[64 * 4];        // conv0 weights, K padded 3 -> 4

  const int tid  = threadIdx.x;
  const int lane = tid & 31;
  const int wv   = __builtin_amdgcn_readfirstlane(tid >> 5);
  const int cs0  = blockIdx.x * NC;

  // ---- stage zero-padded conv0 weights (removes guards from the K-loop) ----
  if (tid < 64) {
    s_w0[tid * 4 + 0] = P.w0[tid * 3 + 0];
    s_w0[tid * 4 + 1] = P.w0[tid * 3 + 1];
    s_w0[tid * 4 + 2] = P.w0[tid * 3 + 2];
    s_w0[tid * 4 + 3] = 0.0f;
  }
  // ---- gather: g = x[knn] - centroid, K padded 3 -> 4 ----
  if (tid < 64) {
    const int ci = tid / KP;
    const int j  = tid % KP;
    const int cs = cs0 + ci;
    const int b  = cs >> 10;
    const int idx = knn[(size_t)cs * KMAX + j];
    const float* pv = x + ((size_t)b * N_ + idx) * 3;
    s_a[tid * 4 + 0] = pv[0] - cent[cs * 3 + 0];
    s_a[tid * 4 + 1] = pv[1] - cent[cs * 3 + 1];
    s_a[tid * 4 + 2] = pv[2] - cent[cs * 3 + 2];
    s_a[tid * 4 + 3] = 0.0f;
  }
  __syncthreads();

  // L1: [64x4] x [4x64]
  gemm_bn_relu<4, 4, 4>(s_a, 4, s_w0, 4,
                        P.b0, P.g0, P.be0, P.mu0, P.va0,
                        s_b, LD1, wv, lane);
  __syncthreads();
  // L2: [64x64] x [64x64]
  gemm_bn_relu<4, 4, 64>(s_b, LD1, P.w1, 64,
                         P.b1, P.g1, P.be1, P.mu1, P.va1,
                         s_a, LD1, wv, lane);
  __syncthreads();
  // L3: [64x64] x [64x128]
  gemm_bn_relu<4, 8, 64>(s_a, LD1, P.w2, 64,
                         P.b2, P.g2, P.be2, P.mu2, P.va2,
                         s_b, LD2, wv, lane);
  __syncthreads();

  // ---- max-pool over KP points -> h (rows 0..NC-1), zero-pad to 16 rows ----
  for (int t = tid; t < 16 * 128; t += 256) {
    const int m  = t >> 7;
    const int ch = t & 127;
    float v = 0.0f;                      // post-ReLU values are >= 0
    if (m < NC)
      for (int r = 0; r < KP; ++r)
        v = fmaxf(v, s_b[(m * KP + r) * LD2 + ch]);
    s_a[m * LD2 + ch] = v;
  }
  __syncthreads();

  // R1: h[16x128] x [128x128] -> r1 at s_a + 16*LD2
  gemm_bn_relu<1, 8, 128>(s_a, LD2, P.rw1, 128,
                          P.rb1, P.rg1, P.rbe1, P.rmu1, P.rva1,
                          s_a + 16 * LD2, LD2, wv, lane);
  __syncthreads();

  // R2 + residual + ReLU + global store (8 N-tiles, one per wave)
  {
    const int rowSel = lane & 15;
    const int kSel   = (lane >> 4) << 1;
    const int nt = wv;
    v8f acc = {};
    const float* ap = s_a + 16 * LD2 + rowSel * LD2 + kSel;
    const float* bp = P.rw2 + (nt * 16 + rowSel) * 128 + kSel;
#pragma unroll
    for (int k0 = 0; k0 < 128; k0 += 4) {
      v2f a, bm;
      a[0]  = ap[k0 + 0];
      a[1]  = ap[k0 + 1];
      bm[0] = bp[k0 + 0];
      bm[1] = bp[k0 + 1];
      acc = __builtin_amdgcn_wmma_f32_16x16x4_f32(
          false, a, false, bm, (short)0, acc, false, false);
    }
    const int n = nt * 16 + rowSel;
    const float sc = P.rg2[n] * rsqrtf(P.rva2[n] + BN_EPS);
    const float sh = P.rbe2[n] - P.rmu2[n] * sc;
    const float bb = P.rb2[n];
    const int m0 = (lane >> 4) << 3;
#pragma unroll
    for (int i = 0; i < 8; ++i) {
      const int m = m0 + i;
      if (m < NC) {
        const float v = (acc[i] + bb) * sc + sh;
        const float h = s_a[m * LD2 + n];
        const size_t cs = (size_t)(cs0 + m);
        out[cs * OUTC + cbase + n] = fmaxf(v + h, 0.0f);
      }
    }
  }
}

// ---------------------------------------------------------------------------
extern "C" void kernel_launch(void* const* d_in, const int* in_sizes, int n_in,
                              void* d_out, int out_size, void* d_ws,
                              size_t ws_size, hipStream_t stream)
{
  const float* x = (const float*)d_in[0];
  MlpParams P;
  P.w0  = (const float*)d_in[1];  P.b0   = (const float*)d_in[2];
  P.g0  = (const float*)d_in[3];  P.be0  = (const float*)d_in[4];
  P.mu0 = (const float*)d_in[5];  P.va0  = (const float*)d_in[6];
  P.w1  = (const float*)d_in[7];  P.b1   = (const float*)d_in[8];
  P.g1  = (const float*)d_in[9];  P.be1  = (const float*)d_in[10];
  P.mu1 = (const float*)d_in[11]; P.va1  = (const float*)d_in[12];
  P.w2  = (const float*)d_in[13]; P.b2   = (const float*)d_in[14];
  P.g2  = (const float*)d_in[15]; P.be2  = (const float*)d_in[16];
  P.mu2 = (const float*)d_in[17]; P.va2  = (const float*)d_in[18];
  P.rw1 = (const float*)d_in[19]; P.rb1  = (const float*)d_in[20];
  P.rg1 = (const float*)d_in[21]; P.rbe1 = (const float*)d_in[22];
  P.rmu1= (const float*)d_in[23]; P.rva1 = (const float*)d_in[24];
  P.rw2 = (const float*)d_in[25]; P.rb2  = (const float*)d_in[26];
  P.rg2 = (const float*)d_in[27]; P.rbe2 = (const float*)d_in[28];
  P.rmu2= (const float*)d_in[29]; P.rva2 = (const float*)d_in[30];

  float* centw = (float*)d_ws;                                  // B*S*3 f32
  int*   knnw  = (int*)((char*)d_ws + sizeof(float) * B_ * S_ * 3); // B*S*32
  float* outp  = (float*)d_out;

  fps_kernel<<<B_, 1024, 0, stream>>>(x, centw, outp);
  knn_kernel<<<(B_ * S_) / 8, 256, 0, stream>>>(x, centw, knnw);
  mlp_kernel<16><<<(B_ * S_) / 4, 256, 0, stream>>>(x, centw, knnw, P, outp, 3);
  mlp_kernel<32><<<(B_ * S_) / 2, 256, 0, stream>>>(x, centw, knnw, P, outp, 3 + 128);
}